// InstanceBank_66898410602530
// MI455X (gfx1250) — compile-verified
//
#include <hip/hip_runtime.h>
#include <stdint.h>

#define BS 64
#define NA 900
#define NT 600
#define ED 256
#define AD 11
#define NC 10
#define NSEL (NA - NT)   // 300
#define MD (ED + AD)     // 267
#define CONF_DECAY 0.6f

// ---------------------------------------------------------------------------
// CDNA5 (gfx1250) async global<->LDS helpers. Inline asm per
// CDNA5 ISA (portable across ROCm7.2 / amdgpu-toolchain).
// Each lane supplies its own global address (VGPR pair) and LDS byte offset
// (VGPR); completion tracked with ASYNCcnt, waited per-wave.
// ---------------------------------------------------------------------------
__device__ __forceinline__ uint32_t lds_off(const void* p) {
  // Low 32 bits of a generic pointer into LDS == LDS byte address (aperture
  // scheme).
  return (uint32_t)(uintptr_t)p;
}

__device__ __forceinline__ void async_g2l_b32(uint32_t lds, const float* g) {
  asm volatile("global_load_async_to_lds_b32 %0, %1, off"
               :: "v"(lds), "v"((uint64_t)(uintptr_t)g) : "memory");
}

__device__ __forceinline__ void async_l2g_b32(float* g, uint32_t lds) {
  asm volatile("global_store_async_from_lds_b32 %0, %1, off"
               :: "v"((uint64_t)(uintptr_t)g), "v"(lds) : "memory");
}

__device__ __forceinline__ void wait_async0() {
  asm volatile("s_wait_asynccnt 0" ::: "memory");
}

// ---------------------------------------------------------------------------
// Kernel 1: one block per batch. Computes conf_max, conf_c (sigmoid + decay
// blend), then exact stable-descending-sort ranks for both top-k selections
// (rank_i = #{v_j > v_i} + #{j < i : v_j == v_i}, matching jax.lax.top_k tie
// rules). Writes idx1 (top-300 of conf_max), idx2 (top-600 of conf_c) and
// new_conf (sorted top-600 values).
// ---------------------------------------------------------------------------
__global__ __launch_bounds__(1024) void ib_topk(
    const float* __restrict__ confidence, const float* __restrict__ prev_conf,
    int* __restrict__ idx1, int* __restrict__ idx2,
    float* __restrict__ new_conf) {
  __shared__ float smax[NA];
  __shared__ float sconf[NA];
  const int b = blockIdx.x;
  const int i = threadIdx.x;
  float vmax = 0.f, vc = 0.f;
  if (i < NA) {
    const float* cp = confidence + ((size_t)b * NA + i) * NC;
    float m = cp[0];
#pragma unroll
    for (int c = 1; c < NC; ++c) m = fmaxf(m, cp[c]);
    float s = 1.0f / (1.0f + expf(-m));
    if (i < NT) s = fmaxf(prev_conf[(size_t)b * NT + i] * CONF_DECAY, s);
    smax[i] = m;
    sconf[i] = s;
    vmax = m;
    vc = s;
  }
  __syncthreads();
  if (i < NA) {
    int r1 = 0, r2 = 0;
    for (int j = 0; j < NA; ++j) {   // broadcast LDS reads: conflict-free
      const float a = smax[j];
      const float c = sconf[j];
      r1 += (a > vmax) || ((a == vmax) && (j < i));
      r2 += (c > vc) || ((c == vc) && (j < i));
    }
    if (r1 < NSEL) idx1[b * NSEL + r1] = i;
    if (r2 < NT) {
      idx2[b * NT + r2] = i;
      new_conf[(size_t)b * NT + r2] = vc;
    }
  }
}

// ---------------------------------------------------------------------------
// Kernel 2: one 256-thread block per merged row (b, i). Selects the feature /
// anchor source row (cached vs top-300 gather vs passthrough, per mask) and
// copies the 256-float feature slice via the async global->LDS->global path
// (B32 per lane; merged rows are 267 floats => only 4B-aligned). Anchor tail
// (11 floats) copied directly.
// ---------------------------------------------------------------------------
__global__ __launch_bounds__(ED) void ib_build_merged(
    const float* __restrict__ inst_f, const float* __restrict__ anchor,
    const float* __restrict__ cached_f, const float* __restrict__ cached_a,
    const int* __restrict__ mask, const int* __restrict__ idx1,
    float* __restrict__ merged) {
  __shared__ float stage[ED];
  const int row = blockIdx.x;  // b*NA + i
  const int b = row / NA;
  const int i = row - b * NA;
  const int t = threadIdx.x;
  const bool m = mask[b] != 0;

  const float* fsrc;
  const float* asrc;
  if (m) {
    if (i < NT) {
      fsrc = cached_f + ((size_t)b * NT + i) * ED;
      asrc = cached_a + ((size_t)b * NT + i) * AD;
    } else {
      const int s = idx1[b * NSEL + (i - NT)];
      fsrc = inst_f + ((size_t)b * NA + s) * ED;
      asrc = anchor + ((size_t)b * NA + s) * AD;
    }
  } else {
    fsrc = inst_f + ((size_t)b * NA + i) * ED;
    asrc = anchor + ((size_t)b * NA + i) * AD;
  }

  float* dst = merged + (size_t)row * MD;
  const uint32_t sl = lds_off(&stage[t]);
  async_g2l_b32(sl, fsrc + t);   // each lane owns its LDS slot (same wave)
  wait_async0();
  async_l2g_b32(dst + t, sl);
  if (t < AD) dst[ED + t] = asrc[t];
}

// ---------------------------------------------------------------------------
// Kernel 3: one 256-thread block per output row (b, r). Gathers row
// i = idx2[b][r] from merged (61.5 MB -> resident in 192 MB L2) into cf/ca,
// again through the async LDS path.
// ---------------------------------------------------------------------------
__global__ __launch_bounds__(ED) void ib_gather(
    const float* __restrict__ merged, const int* __restrict__ idx2,
    float* __restrict__ cf, float* __restrict__ ca) {
  __shared__ float stage[ED];
  const int row = blockIdx.x;  // b*NT + r
  const int b = row / NT;
  const int r = row - b * NT;
  const int t = threadIdx.x;
  const int i = idx2[b * NT + r];
  const float* src = merged + ((size_t)b * NA + i) * MD;

  const uint32_t sl = lds_off(&stage[t]);
  async_g2l_b32(sl, src + t);
  wait_async0();
  async_l2g_b32(cf + (size_t)row * ED + t, sl);
  if (t < AD) ca[(size_t)row * AD + t] = src[ED + t];
}

// ---------------------------------------------------------------------------
extern "C" void kernel_launch(void* const* d_in, const int* in_sizes, int n_in,
                              void* d_out, int out_size, void* d_ws,
                              size_t ws_size, hipStream_t stream) {
  const float* inst_f   = (const float*)d_in[0];  // (BS, NA, ED)
  const float* anchor   = (const float*)d_in[1];  // (BS, NA, AD)
  const float* conf     = (const float*)d_in[2];  // (BS, NA, NC)
  const float* cached_f = (const float*)d_in[3];  // (BS, NT, ED)
  const float* cached_a = (const float*)d_in[4];  // (BS, NT, AD)
  const float* prev_c   = (const float*)d_in[5];  // (BS, NT)
  const int*   mask     = (const int*)d_in[6];    // (BS,) bool as int

  float* out      = (float*)d_out;
  float* merged   = out;                                   // BS*NA*MD
  float* new_conf = out + (size_t)BS * NA * MD;            // BS*NT
  float* cf       = new_conf + (size_t)BS * NT;            // BS*NT*ED
  float* ca       = cf + (size_t)BS * NT * ED;             // BS*NT*AD

  int* idx1 = (int*)d_ws;            // BS*NSEL
  int* idx2 = idx1 + BS * NSEL;      // BS*NT   (total 230.4 KB scratch)

  ib_topk<<<BS, 1024, 0, stream>>>(conf, prev_c, idx1, idx2, new_conf);
  ib_build_merged<<<BS * NA, ED, 0, stream>>>(inst_f, anchor, cached_f,
                                              cached_a, mask, idx1, merged);
  ib_gather<<<BS * NT, ED, 0, stream>>>(merged, idx2, cf, ca);
}